// Group_50096498541038
// MI455X (gfx1250) — compile-verified
//
#include <hip/hip_runtime.h>
#include <hip/hip_bf16.h>
#include <math.h>

#define B_  16
#define N_  8192
#define NG  512
#define GS  32
#define TILE 256

typedef __attribute__((ext_vector_type(2))) float v2f;
typedef __attribute__((ext_vector_type(8))) float v8f;

// argmax over the wave of (v, i); ties -> lower i. All lanes converge to result.
__device__ __forceinline__ void waveArgMax(float v, int i, float& mv, int& mi) {
    mv = v; mi = i;
#pragma unroll
    for (int off = 16; off > 0; off >>= 1) {
        float ov = __shfl_xor(mv, off);
        int   oi = __shfl_xor(mi, off);
        if (ov > mv || (ov == mv && oi < mi)) { mv = ov; mi = oi; }
    }
}

// max over the wave of v, returning value and owning lane (ties -> lower lane).
__device__ __forceinline__ void waveMaxLane(float v, float& mv, int& ml) {
    mv = v; ml = (int)(threadIdx.x & 31);
#pragma unroll
    for (int off = 16; off > 0; off >>= 1) {
        float ov = __shfl_xor(mv, off);
        int   ol = __shfl_xor(ml, off);
        if (ov > mv || (ov == mv && ol < ml)) { mv = ov; ml = ol; }
    }
}

// ---------------------------------------------------------------------------
// Kernel 1: farthest point sampling. One workgroup (32 waves) per batch.
// Each thread owns 8 points entirely in registers (xyz + running min-dist).
// Per step: broadcast last pick via LDS, VALU distance update, two-level
// shuffle/LDS argmax reduction. 511 dependent steps -> latency bound, so all
// state stays on-chip.
// ---------------------------------------------------------------------------
__global__ __launch_bounds__(1024) void fps_kernel(const float* __restrict__ pts,
                                                   float* __restrict__ center) {
    const int b    = blockIdx.x;
    const int tid  = threadIdx.x;
    const int lane = tid & 31;
    const int wave = tid >> 5;

    __shared__ float bc[3];
    __shared__ float redV[32];
    __shared__ int   redI[32];
    __shared__ int   lastSh;

    const float* base = pts + (size_t)b * N_ * 6;

    float px[8], py[8], pz[8], dist[8];
#pragma unroll
    for (int k = 0; k < 8; ++k) {
        int n = tid + k * 1024;
        px[k] = base[n * 6 + 0];
        py[k] = base[n * 6 + 1];
        pz[k] = base[n * 6 + 2];
        dist[k] = INFINITY;
    }

    int last = 0;
    for (int i = 0; i < NG; ++i) {
        if (tid == (last & 1023)) {
            int k = last >> 10;
            bc[0] = px[k]; bc[1] = py[k]; bc[2] = pz[k];
            float* c = center + ((size_t)b * NG + i) * 3;
            c[0] = px[k]; c[1] = py[k]; c[2] = pz[k];
        }
        __syncthreads();
        if (i == NG - 1) break;

        const float lx = bc[0], ly = bc[1], lz = bc[2];
        float bv = -1.0f; int bi = 0;
#pragma unroll
        for (int k = 0; k < 8; ++k) {
            float dx = px[k] - lx, dy = py[k] - ly, dz = pz[k] - lz;
            float d  = dx * dx + dy * dy + dz * dz;
            d = fminf(dist[k], d);
            dist[k] = d;
            if (d > bv) { bv = d; bi = tid + k * 1024; }
        }
        float mv; int mi;
        waveArgMax(bv, bi, mv, mi);
        if (lane == 0) { redV[wave] = mv; redI[wave] = mi; }
        __syncthreads();
        if (wave == 0) {
            float mv2; int mi2;
            waveArgMax(redV[lane], redI[lane], mv2, mi2);
            if (lane == 0) lastSh = mi2;
        }
        __syncthreads();
        last = lastSh;
    }
}

// ---------------------------------------------------------------------------
// Kernel 2: fused distance-map + top-32 + gather. One workgroup per
// (batch, 16-center tile); 8 waves. s[g,n] = -2 c.x - ... + |p|^2 is computed
// as A(16x4) x B(4x16) with V_WMMA_F32_16X16X4_F32. LDS tiles are laid out as
// 4 contiguous planes so A/B fragments load with one cndmask'd plane offset
// and unconditional ds_loads (no EXEC diamonds around the WMMA). Each wave
// then owns 2 centers' top-32 lists (one (val,idx) per lane) with a ballot
// threshold filter so almost all candidates cost ~2 VALU ops.
// ---------------------------------------------------------------------------
__global__ __launch_bounds__(256) void group_kernel(const float* __restrict__ pts,
                                                    const float* __restrict__ center,
                                                    float* __restrict__ outNb) {
    const int wg    = blockIdx.x;           // B_ * 32 workgroups
    const int b     = wg >> 5;
    const int gbase = (wg & 31) * 16;
    const int tid   = threadIdx.x;
    const int lane  = tid & 31;
    const int wave  = tid >> 5;
    const int lm    = lane & 15;

    __shared__ float cpl[4 * 16];       // planes: -2cx | -2cy | -2cz | 1
    __shared__ float tp[4 * TILE];      // planes:   x  |   y  |   z  | |p|^2
    __shared__ float sT[16 * TILE];

    if (tid < 16) {
        const float* c = center + ((size_t)b * NG + gbase + tid) * 3;
        cpl[0 * 16 + tid] = -2.0f * c[0];
        cpl[1 * 16 + tid] = -2.0f * c[1];
        cpl[2 * 16 + tid] = -2.0f * c[2];
        cpl[3 * 16 + tid] = 1.0f;
    }
    __syncthreads();

    // A fragment (16x4 f32): lanes 0-15 hold K=0/1, lanes 16-31 hold K=2/3.
    const int aplane = (lane < 16) ? 0 : 2;       // cndmask, no branch
    v2f A;
    A.x = cpl[aplane * 16 + lm];
    A.y = cpl[(aplane + 1) * 16 + lm];

    // Branchless B-fragment base offset within the plane array (loop-invariant
    // except for the subtile column base).
    const int bplane = (lane < 16) ? 0 : 2;
    const int boff   = bplane * TILE + lm;
    const int rowoff = (lane < 16) ? 0 : 8;

    float bv[2]    = {INFINITY, INFINITY};
    int   bi[2]    = {0, 0};
    float worst[2] = {INFINITY, INFINITY};
    const int m0 = wave * 2;

    const float* pbase = pts + (size_t)b * N_ * 6;

    for (int t = 0; t < N_ / TILE; ++t) {
        const int nbase = t * TILE;
        {
            int n = nbase + tid;
            float x = pbase[n * 6 + 0];
            float y = pbase[n * 6 + 1];
            float z = pbase[n * 6 + 2];
            tp[0 * TILE + tid] = x;
            tp[1 * TILE + tid] = y;
            tp[2 * TILE + tid] = z;
            tp[3 * TILE + tid] = x * x + y * y + z * z;
            if (t + 1 < N_ / TILE)
                __builtin_prefetch(&pbase[(n + TILE) * 6], 0, 0);  // global_prefetch_b8
        }
        __syncthreads();

        // 16 N-subtiles of 16 points; this wave computes subtiles 2w, 2w+1.
#pragma unroll
        for (int s = 0; s < 2; ++s) {
            const int cbase = (wave * 2 + s) * 16;
            v2f Bm;
            Bm.x = tp[boff + cbase];
            Bm.y = tp[boff + cbase + TILE];
            v8f Cz = {};
            v8f D = __builtin_amdgcn_wmma_f32_16x16x4_f32(
                false, A, false, Bm, (short)0, Cz, false, false);
            const int sbase = rowoff * TILE + cbase + lm;
#pragma unroll
            for (int r = 0; r < 8; ++r)
                sT[r * TILE + sbase] = D[r];
        }
        __syncthreads();

        // Threshold-filtered top-32 merge for this wave's two centers.
#pragma unroll
        for (int s = 0; s < 2; ++s) {
            const int m = m0 + s;
            for (int bb = 0; bb < TILE / 32; ++bb) {
                float cval = sT[m * TILE + bb * 32 + lane];
                int   cidx = nbase + bb * 32 + lane;
                unsigned long long mask = __ballot(cval < worst[s]);
                while (mask) {
                    int j = (int)__ffsll(mask) - 1;
                    mask &= mask - 1;
                    float cv = __shfl(cval, j);
                    int   cn = __shfl(cidx, j);
                    float mv; int ml;
                    waveMaxLane(bv[s], mv, ml);
                    if (cv < mv && lane == ml) { bv[s] = cv; bi[s] = cn; }
                    worst[s] = mv;   // max only decreases -> conservative
                }
            }
        }
        // No barrier needed here: tile t+1's post-load barrier orders these
        // sT reads before tile t+1's sT writes.
    }
    __syncthreads();

    // Sort each center's 32 survivors ascending (value, then index) across
    // lanes via bitonic network, then gather/recenter to the output.
#pragma unroll
    for (int s = 0; s < 2; ++s) {
        float v  = bv[s];
        int   ix = bi[s];
#pragma unroll
        for (int k = 2; k <= 32; k <<= 1) {
#pragma unroll
            for (int j = k >> 1; j >= 1; j >>= 1) {
                float ov = __shfl_xor(v, j);
                int   oi = __shfl_xor(ix, j);
                bool up      = ((lane & k) == 0);
                bool lower   = ((lane & j) == 0);
                bool takeMin = (up == lower);
                bool gt = (v > ov) || (v == ov && ix > oi);
                bool sel = takeMin ? gt : !gt;
                if (sel) { v = ov; ix = oi; }
            }
        }
        const int m = m0 + s;
        const int g = gbase + m;
        const float* p = pbase + (size_t)ix * 6;
        float o0 = p[0] + 0.5f * cpl[0 * 16 + m];   // p - c  (plane holds -2c)
        float o1 = p[1] + 0.5f * cpl[1 * 16 + m];
        float o2 = p[2] + 0.5f * cpl[2 * 16 + m];
        float o3 = p[3], o4 = p[4], o5 = p[5];
        float* ob = outNb + ((size_t)(b * NG + g) * GS + lane) * 6;
        ob[0] = o0; ob[1] = o1; ob[2] = o2;
        ob[3] = o3; ob[4] = o4; ob[5] = o5;
    }
}

extern "C" void kernel_launch(void* const* d_in, const int* in_sizes, int n_in,
                              void* d_out, int out_size, void* d_ws, size_t ws_size,
                              hipStream_t stream) {
    const float* pts = (const float*)d_in[0];
    float* out    = (float*)d_out;
    float* nb     = out;                                   // (B, 512, 32, 6)
    float* center = out + (size_t)B_ * NG * GS * 6;        // (B, 512, 3)

    fps_kernel<<<B_, 1024, 0, stream>>>(pts, center);
    group_kernel<<<B_ * 32, 256, 0, stream>>>(pts, center, nb);
}